// GCN_65704409694465
// MI455X (gfx1250) — compile-verified
//
#include <hip/hip_runtime.h>

typedef float v2f __attribute__((ext_vector_type(2)));
typedef float v8f __attribute__((ext_vector_type(8)));

// ---------------------------------------------------------------------------
// One-time degree / normalization kernels
// ---------------------------------------------------------------------------
__global__ void deg_init_k(float* __restrict__ deg, int n) {
    int i = blockIdx.x * blockDim.x + threadIdx.x;
    if (i < n) deg[i] = 1.0f;                       // self-loop
}

__global__ void deg_count_k(const int* __restrict__ dst, float* __restrict__ deg, int e) {
    int i = blockIdx.x * blockDim.x + threadIdx.x;
    if (i < e) atomicAdd(&deg[dst[i]], 1.0f);
}

__global__ void deg_rsqrt_k(float* __restrict__ deg, int n) {
    int i = blockIdx.x * blockDim.x + threadIdx.x;
    if (i < n) deg[i] = rsqrtf(deg[i]);
}

// ---------------------------------------------------------------------------
// Pack W [K][NC] row-major -> Wp [K/2][NC] of float2 {W[2k][c], W[2k+1][c]}
// so a WMMA B-fragment is one 8-byte load per lane per K-step.
// ---------------------------------------------------------------------------
__global__ void pack_w_k(const float* __restrict__ W, v2f* __restrict__ Wp, int K, int NC) {
    int i = blockIdx.x * blockDim.x + threadIdx.x;
    int total = (K / 2) * NC;
    if (i >= total) return;
    int kk = i / NC, c = i % NC;
    v2f p;
    p.x = W[(size_t)(2 * kk) * NC + c];
    p.y = W[(size_t)(2 * kk + 1) * NC + c];
    Wp[i] = p;
}

// ---------------------------------------------------------------------------
// GEMM: h = X @ W  (fp32, V_WMMA_F32_16X16X4_F32), fused epilogue writes
//   h_out  = h * dinv          (pre-scaled: edge term becomes hs[src]*dinv[dst])
//   agg    = h * dinv^2 + bias (self-loop + bias init for the scatter)
// Block: 16-row tile (staged in LDS), NC/16 waves, one 16-col tile per wave.
// ---------------------------------------------------------------------------
template <int NC>
__global__ __launch_bounds__(NC * 2) void gcn_gemm_k(
    const float* __restrict__ X, const v2f* __restrict__ Wp,
    const float* __restrict__ bias, const float* __restrict__ dinv,
    float* __restrict__ h_out, float* __restrict__ agg_out, int nrows)
{
    constexpr int K = 128;
    constexpr int LSTR = K + 4;                     // pad to dodge bank conflicts
    __shared__ float Xs[16 * LSTR];

    const int tid = threadIdx.x;
    const int row0 = blockIdx.x * 16;

    // cooperative, coalesced stage of the 16x128 X tile
    for (int i = tid; i < 16 * K; i += NC * 2) {
        int r = i >> 7, k = i & 127;
        int gr = row0 + r;
        Xs[r * LSTR + k] = (gr < nrows) ? X[(size_t)gr * K + k] : 0.0f;
    }
    __syncthreads();

    const int wave = tid >> 5;                      // column tile
    const int lane = tid & 31;
    const int half = lane >> 4;                     // lane group 0/1
    const int l16  = lane & 15;
    const int col  = wave * 16 + l16;

    v8f acc = {};
    const float* xrow = &Xs[l16 * LSTR];
#pragma unroll
    for (int k = 0; k < K; k += 4) {
        const int ka = k + half * 2;                // lanes 0-15: K=k,k+1 ; 16-31: K=k+2,k+3
        v2f a;
        a.x = xrow[ka];
        a.y = xrow[ka + 1];
        const v2f b = Wp[(size_t)((k >> 1) + half) * NC + col];
        acc = __builtin_amdgcn_wmma_f32_16x16x4_f32(
            /*neg_a=*/false, a, /*neg_b=*/false, b,
            /*c_mod=*/(short)0, acc, /*reuse_a=*/false, /*reuse_b=*/false);
    }

    const float bv = bias[col];
#pragma unroll
    for (int i = 0; i < 8; i++) {
        const int m = half * 8 + i;                 // C/D layout: VGPR i -> row i / 8+i
        const int r = row0 + m;
        if (r < nrows) {
            const float di = dinv[r];
            const float hs = acc[i] * di;           // pre-scaled h
            const size_t off = (size_t)r * NC + col;
            h_out[off] = hs;
            agg_out[off] = hs * di + bv;            // h*dinv^2 + b
        }
    }
}

// ---------------------------------------------------------------------------
// Edge scatter: agg[dst] += dinv[dst] * hs[src].  One wave per edge (D=128):
// edge metadata scalarized (wave-uniform -> SMEM loads), lanes cover the row
// as float4 chunks -> 512B coalesced gather, atomics to consecutive
// addresses (output is L2-resident: 51MB << 192MB).
// ---------------------------------------------------------------------------
template <int D>
__global__ void gcn_scatter_k(const float* __restrict__ h, const int* __restrict__ src,
                              const int* __restrict__ dst, const float* __restrict__ dinv,
                              float* __restrict__ out, long long total)
{
    constexpr int CH = D / 4;
    long long idx = (long long)blockIdx.x * blockDim.x + threadIdx.x;
    if (idx >= total) return;
    int e = (int)(idx / CH);
    const int c = (int)(idx % CH);
    if constexpr (CH == 32) {
        // one edge per wave; total is a multiple of 32 so the guard above is
        // wave-granular and e is wave-uniform -> scalar loads
        e = __builtin_amdgcn_readfirstlane(e);
    }
    const int s = src[e];
    const int d = dst[e];
    const float w = dinv[d];
    const float4 v = ((const float4*)(h + (size_t)s * D))[c];
    float* o = out + (size_t)d * D + (size_t)c * 4;
    atomicAdd(o + 0, v.x * w);
    atomicAdd(o + 1, v.y * w);
    atomicAdd(o + 2, v.z * w);
    atomicAdd(o + 3, v.w * w);
}

__global__ void relu_k(float* __restrict__ x, long long n) {
    long long i = (long long)blockIdx.x * blockDim.x + threadIdx.x;
    if (i < n) x[i] = fmaxf(x[i], 0.0f);
}

// ---------------------------------------------------------------------------
// Driver
// ---------------------------------------------------------------------------
extern "C" void kernel_launch(void* const* d_in, const int* in_sizes, int n_in,
                              void* d_out, int out_size, void* d_ws, size_t ws_size,
                              hipStream_t stream) {
    const float* x  = (const float*)d_in[0];
    const int*   ei = (const int*)d_in[1];
    const float* W0 = (const float*)d_in[2];
    const float* b0 = (const float*)d_in[3];
    const float* W1 = (const float*)d_in[4];
    const float* b1 = (const float*)d_in[5];
    const float* W2 = (const float*)d_in[6];
    const float* b2 = (const float*)d_in[7];
    float* out = (float*)d_out;

    const int n = in_sizes[0] / 128;
    const int e = in_sizes[1] / 2;
    const int* src = ei;
    const int* dst = ei + e;

    float* ws = (float*)d_ws;
    const size_t nd = (size_t)n * 128;
    float* hbuf = ws;               // n*128
    float* actA = ws + nd;          // n*128
    float* actB = ws + 2 * nd;      // n*128
    float* dinv = ws + 3 * nd;      // n   (holds deg, then deg^-1/2)
    size_t woff = 3 * nd + (((size_t)n + 1) & ~(size_t)1);   // 8B aligned
    v2f* Wp0 = (v2f*)(ws + woff);   // 64*128 float2
    v2f* Wp1 = Wp0 + 64 * 128;      // 64*128 float2
    v2f* Wp2 = Wp1 + 64 * 128;      // 64*64  float2

    const int nb_n = (n + 255) / 256;
    const int nb_e = (e + 255) / 256;
    deg_init_k<<<nb_n, 256, 0, stream>>>(dinv, n);
    deg_count_k<<<nb_e, 256, 0, stream>>>(dst, dinv, e);
    deg_rsqrt_k<<<nb_n, 256, 0, stream>>>(dinv, n);

    pack_w_k<<<(64 * 128 + 255) / 256, 256, 0, stream>>>(W0, Wp0, 128, 128);
    pack_w_k<<<(64 * 128 + 255) / 256, 256, 0, stream>>>(W1, Wp1, 128, 128);
    pack_w_k<<<(64 * 64 + 255) / 256, 256, 0, stream>>>(W2, Wp2, 128, 64);

    const int rtiles = (n + 15) / 16;
    const long long tot128 = (long long)e * 32;
    const long long tot64  = (long long)e * 16;
    const int sb128 = (int)((tot128 + 255) / 256);
    const int sb64  = (int)((tot64 + 255) / 256);
    const long long ndll = (long long)nd;
    const int rb = (int)((ndll + 255) / 256);

    // Layer 0: x -> actA
    gcn_gemm_k<128><<<rtiles, 256, 0, stream>>>(x, Wp0, b0, dinv, hbuf, actA, n);
    gcn_scatter_k<128><<<sb128, 256, 0, stream>>>(hbuf, src, dst, dinv, actA, tot128);
    relu_k<<<rb, 256, 0, stream>>>(actA, ndll);

    // Layer 1: actA -> actB
    gcn_gemm_k<128><<<rtiles, 256, 0, stream>>>(actA, Wp1, b1, dinv, hbuf, actB, n);
    gcn_scatter_k<128><<<sb128, 256, 0, stream>>>(hbuf, src, dst, dinv, actB, tot128);
    relu_k<<<rb, 256, 0, stream>>>(actB, ndll);

    // Layer 2: actB -> d_out (D_OUT = 64)
    gcn_gemm_k<64><<<rtiles, 128, 0, stream>>>(actB, Wp2, b2, dinv, hbuf, out, n);
    gcn_scatter_k<64><<<sb64, 256, 0, stream>>>(hbuf, src, dst, dinv, out, tot64);
}